// EmbedderGNNv2_46445776339648
// MI455X (gfx1250) — compile-verified
//
#include <hip/hip_runtime.h>
#include <hip/hip_bf16.h>

typedef __attribute__((ext_vector_type(2))) float v2f;
typedef __attribute__((ext_vector_type(4))) float v4f;
typedef __attribute__((ext_vector_type(8))) float v8f;

#define D 128

// ---------------- utility kernels ----------------

__global__ void fill_zero(float* __restrict__ p, unsigned n) {
  unsigned i = blockIdx.x * blockDim.x + threadIdx.x;
  unsigned stride = gridDim.x * blockDim.x;
  for (; i < n; i += stride) p[i] = 0.0f;
}

__global__ void count_edges(const int* __restrict__ ei, float* __restrict__ cnt, int E) {
  int e = blockIdx.x * blockDim.x + threadIdx.x;
  if (e >= E) return;
  int dst = ei[E + e];
  atomicAdd(&cnt[dst], 1.0f);
}

__global__ void inv_counts(const float* __restrict__ cnt, float* __restrict__ inv, int n) {
  int i = blockIdx.x * blockDim.x + threadIdx.x;
  if (i >= n) return;
  inv[i] = 1.0f / fmaxf(cnt[i], 1.0f);
}

// one thread per (edge, 4-feature chunk): 32 threads/edge
__global__ void scatter_add(const float* __restrict__ hin, const int* __restrict__ ei,
                            float* __restrict__ agg, int E) {
  unsigned tid = blockIdx.x * blockDim.x + threadIdx.x;
  unsigned total = (unsigned)E * 32u;
  if (tid >= total) return;
  int e = tid >> 5;
  int c = (tid & 31) << 2;
  int src = ei[e];
  int dst = ei[E + e];
  v4f v = *(const v4f*)(hin + (size_t)src * D + c);
  float* a = agg + (size_t)dst * D + c;
  atomicAdd(a + 0, v.x);
  atomicAdd(a + 1, v.y);
  atomicAdd(a + 2, v.z);
  atomicAdd(a + 3, v.w);
}

// ---------------- WMMA GEMM: y = (agg*inv) @ Wl^T + h @ Wr^T + bl ----------------
// One wave per 16x16 output tile; K=128 via 32+32 V_WMMA_F32_16X16X4_F32 ops.
// Fused BN column statistics (sum, sumsq) via shfl_xor + atomics.
__global__ void __launch_bounds__(128)
sage_gemm_wmma(const float* __restrict__ agg, const float* __restrict__ inv,
               const float* __restrict__ hin,
               const float* __restrict__ Wl, const float* __restrict__ bl,
               const float* __restrict__ Wr,
               float* __restrict__ y, float* __restrict__ colsum, float* __restrict__ colsq,
               int ntiles_m, int Nn) {
  int lane = threadIdx.x & 31;
  int wave = threadIdx.x >> 5;
  int tile = blockIdx.x * 4 + wave;          // wave-uniform
  int mt = tile >> 3;                        // node-tile (D/16 == 8 feature tiles)
  int nt = tile & 7;
  if (mt >= ntiles_m) return;                // wave-uniform exit: EXEC all-ones for WMMA

  int lr   = lane & 15;                      // row-in-tile (A) / col-in-tile (B,C)
  int half = lane >> 4;                      // K sub-slice select {0,1}<->{2,3}
  int row  = mt * 16 + lr;                   // node row this lane loads for A
  int rowc = row < Nn ? row : (Nn - 1);      // clamp (N=50000 is /16, this is insurance)
  int col  = nt * 16 + lr;                   // output feature column

  float invc = inv[rowc];
  const float* arow  = agg + (size_t)rowc * D + 2 * half;
  const float* hrow  = hin + (size_t)rowc * D + 2 * half;
  const float* wlrow = Wl  + (size_t)col  * D + 2 * half;   // B[k][n] = Wl[col][k]
  const float* wrrow = Wr  + (size_t)col  * D + 2 * half;

  v8f acc = {};
#pragma unroll
  for (int k = 0; k < D; k += 4) {
    v2f a = *(const v2f*)(arow + k);
    a.x *= invc; a.y *= invc;
    v2f b = *(const v2f*)(wlrow + k);
    acc = __builtin_amdgcn_wmma_f32_16x16x4_f32(false, a, false, b, (short)0, acc,
                                                false, false);
  }
#pragma unroll
  for (int k = 0; k < D; k += 4) {
    v2f a = *(const v2f*)(hrow + k);
    v2f b = *(const v2f*)(wrrow + k);
    acc = __builtin_amdgcn_wmma_f32_16x16x4_f32(false, a, false, b, (short)0, acc,
                                                false, false);
  }

  // Epilogue: +bias, store y, accumulate per-column BN partial sums.
  float bias = bl[col];
  float s = 0.0f, s2 = 0.0f;
#pragma unroll
  for (int r = 0; r < 8; ++r) {
    int gr = mt * 16 + r + 8 * half;         // C layout: lanes 0-15 -> M=r, 16-31 -> M=r+8
    float v = acc[r] + bias;
    if (gr < Nn) {
      y[(size_t)gr * D + col] = v;
      s += v;
      s2 += v * v;
    }
  }
  s  += __shfl_xor(s, 16, 32);               // combine the two half-lanes of each column
  s2 += __shfl_xor(s2, 16, 32);
  if (lane < 16) {
    atomicAdd(&colsum[col], s);
    atomicAdd(&colsq[col], s2);
  }
}

// ---------------- BN finalize + apply ----------------

__global__ void bn_finalize(const float* __restrict__ colsum, const float* __restrict__ colsq,
                            const float* __restrict__ gamma, const float* __restrict__ beta,
                            float* __restrict__ scale, float* __restrict__ shift, float invN) {
  int f = threadIdx.x;
  float m  = colsum[f] * invN;
  float vr = colsq[f] * invN - m * m;
  float rs = rsqrtf(vr + 1e-5f);
  float sc = rs * gamma[f];
  scale[f] = sc;
  shift[f] = beta[f] - m * sc;
}

__global__ void bn_apply(const float* __restrict__ y, const float* __restrict__ scale,
                         const float* __restrict__ shift, float* __restrict__ out,
                         unsigned n4, int relu) {
  unsigned i = blockIdx.x * blockDim.x + threadIdx.x;
  if (i >= n4) return;
  int f0 = (i & 31) << 2;                     // feature offset of this float4 (D==128)
  v4f v  = *(const v4f*)(y + (size_t)i * 4);
  v4f sc = *(const v4f*)(scale + f0);
  v4f sh = *(const v4f*)(shift + f0);
  v = v * sc + sh;
  if (relu) {
    v.x = fmaxf(v.x, 0.0f); v.y = fmaxf(v.y, 0.0f);
    v.z = fmaxf(v.z, 0.0f); v.w = fmaxf(v.w, 0.0f);
  }
  *(v4f*)(out + (size_t)i * 4) = v;
}

// ---------------- host launch ----------------

extern "C" void kernel_launch(void* const* d_in, const int* in_sizes, int n_in,
                              void* d_out, int out_size, void* d_ws, size_t ws_size,
                              hipStream_t stream) {
  const float* x  = (const float*)d_in[0];
  const int*   ei = (const int*)d_in[1];     // edge_index [2,E] row-major
  int N = in_sizes[0] / D;
  int E = in_sizes[1] / 2;

  const float* Wl[3]    = {(const float*)d_in[2],  (const float*)d_in[7],  (const float*)d_in[12]};
  const float* bl[3]    = {(const float*)d_in[3],  (const float*)d_in[8],  (const float*)d_in[13]};
  const float* Wr[3]    = {(const float*)d_in[4],  (const float*)d_in[9],  (const float*)d_in[14]};
  const float* gamma[3] = {(const float*)d_in[5],  (const float*)d_in[10], (const float*)d_in[15]};
  const float* beta[3]  = {(const float*)d_in[6],  (const float*)d_in[11], (const float*)d_in[16]};

  float* ws     = (float*)d_ws;
  float* cnt    = ws;  ws += N;
  float* inv    = ws;  ws += N;
  float* agg    = ws;  ws += (size_t)N * D;
  float* ybuf   = ws;  ws += (size_t)N * D;
  float* hbuf   = ws;  ws += (size_t)N * D;
  float* colsum = ws;  ws += D;
  float* colsq  = ws;  ws += D;               // contiguous with colsum (zeroed together)
  float* scale  = ws;  ws += D;
  float* shift  = ws;  ws += D;

  const int TB = 256;

  // Degree counts (shared by all layers)
  fill_zero<<<256, TB, 0, stream>>>(cnt, (unsigned)N);
  count_edges<<<(E + TB - 1) / TB, TB, 0, stream>>>(ei, cnt, E);
  inv_counts<<<(N + TB - 1) / TB, TB, 0, stream>>>(cnt, inv, N);

  int ntiles_m    = (N + 15) / 16;            // 3125 for N=50000
  int tiles       = ntiles_m * (D / 16);
  int gemm_blocks = (tiles + 3) / 4;          // 4 waves (tiles) per 128-thread block

  unsigned nd  = (unsigned)N * D;
  unsigned nd4 = nd / 4;
  unsigned scat_threads = (unsigned)E * 32u;

  float* out = (float*)d_out;
  const float* hin = x;
  for (int l = 0; l < 3; ++l) {
    fill_zero<<<2048, TB, 0, stream>>>(agg, nd);
    fill_zero<<<1, TB, 0, stream>>>(colsum, 2u * D);
    scatter_add<<<(scat_threads + TB - 1) / TB, TB, 0, stream>>>(hin, ei, agg, E);
    sage_gemm_wmma<<<gemm_blocks, 128, 0, stream>>>(agg, inv, hin, Wl[l], bl[l], Wr[l],
                                                    ybuf, colsum, colsq, ntiles_m, N);
    bn_finalize<<<1, D, 0, stream>>>(colsum, colsq, gamma[l], beta[l], scale, shift,
                                     1.0f / (float)N);
    float* hout = (l == 2) ? out : hbuf;
    bn_apply<<<(nd4 + TB - 1) / TB, TB, 0, stream>>>(ybuf, scale, shift, hout, nd4,
                                                     (l < 2) ? 1 : 0);
    hin = hbuf;
  }
}